// nearestneighbor_72679436583161
// MI455X (gfx1250) — compile-verified
//
#include <hip/hip_runtime.h>
#include <hip/hip_bf16.h>

// KNN classifier: dist = 2*(x_test @ x_train^T) - rowsum(x_test) - rowsum(x_train) + D
// Heavy part: 2048 x 50000 x 784 f32 GEMM -> V_WMMA_F32_16X16X4_F32 on gfx1250.
// Working set (163 MB) fits in the 192 MB L2 -> compute/L2 bound.
// This version software-pipelines the K-chunks: GLOBAL_LOAD_ASYNC_TO_LDS_B128
// (ASYNCcnt) prefetches chunk t+1 into the alternate LDS buffer while WMMAs
// consume chunk t; s_wait_asynccnt 0 + barrier flips the buffers.

typedef float v2f __attribute__((ext_vector_type(2)));
typedef float v8f __attribute__((ext_vector_type(8)));

#define NTEST   2048
#define NTRAIN  50000
#define DIM     784
#define NCLS    10

#define ROWS    32          // test rows per block
#define COLS    128         // train rows per block n-tile
#define KC      56          // K chunk (784 = 14*56), 14 float4 per row
#define NKC     (DIM / KC)  // 14
#define NSPLIT  8           // N-dimension splits
#define SPLIT   (NTRAIN / NSPLIT)            // 6250
#define NTILES  ((SPLIT + COLS - 1) / COLS)  // 49
#define NCHUNKS (NTILES * NKC)               // 686

#define LDS_S   60          // LDS row stride (floats): 240B -> 16B-aligned rows,
                            // 4r mod 64 distinct for r<16 -> conflict-free b64 reads
#define FRAME   (ROWS * LDS_S + COLS * LDS_S)   // 9600 floats per pipeline buffer
#define SB_OFF  (ROWS * LDS_S)                  // B tile offset inside a frame
#define SD_OFF  (2 * FRAME)                     // distance staging
#define SD_S    130                             // padded -> conflict-free row scans
#define TOPV_OFF (SD_OFF + ROWS * SD_S)
#define TOPI_OFF (TOPV_OFF + ROWS * 5)
#define SMEM_FLOATS (TOPI_OFF + ROWS * 5)       // 23680 floats = 94720 bytes

#define NEG_INF (-3.402823466e38f)

// ---- gfx1250 async global->LDS copy (ASYNCcnt), per CDNA5 ISA
__device__ __forceinline__ void async_ld128(unsigned ldsByteOff, const float* base,
                                            unsigned gByteOff) {
    asm volatile("global_load_async_to_lds_b128 %0, %1, %2"
                 :
                 : "v"(ldsByteOff), "v"(gByteOff), "s"(base)
                 : "memory");
}
__device__ __forceinline__ void wait_async0() {
    asm volatile("s_wait_asynccnt 0x0" ::: "memory");
}

// Insert (v, idx) into a 5-entry list sorted by (value desc, index asc).
// Matches jax.lax.top_k tie-breaking (smaller index first on equal values).
__device__ __forceinline__ void insert5(float* tv, int* ti, float v, int idx) {
    if (v > tv[4] || (v == tv[4] && idx < ti[4])) {
        int j = 4;
        while (j > 0 && (v > tv[j - 1] || (v == tv[j - 1] && idx < ti[j - 1]))) {
            tv[j] = tv[j - 1];
            ti[j] = ti[j - 1];
            --j;
        }
        tv[j] = v;
        ti[j] = idx;
    }
}

// ---------------------------------------------------------------------------
// Kernel 0: row sums of x_test (2048 rows) and x_train (50000 rows).
// One wave32 per row, coalesced strided reads, shuffle reduction.
// ---------------------------------------------------------------------------
__global__ __launch_bounds__(256) void knn_rowsums(const float* __restrict__ xtest,
                                                   const float* __restrict__ xtrain,
                                                   float* __restrict__ stest,
                                                   float* __restrict__ strain) {
    const int lane = threadIdx.x & 31;
    const int wv   = threadIdx.x >> 5;
    const int row  = blockIdx.x * 8 + wv;   // 8 waves per block
    if (row >= NTEST + NTRAIN) return;

    const float* src;
    float* dst;
    if (row < NTEST) {
        src = xtest + (size_t)row * DIM;
        dst = stest + row;
    } else {
        src = xtrain + (size_t)(row - NTEST) * DIM;
        dst = strain + (row - NTEST);
    }
    float s = 0.f;
    for (int k = lane; k < DIM; k += 32) s += src[k];
    #pragma unroll
    for (int off = 16; off > 0; off >>= 1) s += __shfl_xor(s, off, 32);
    if (lane == 0) *dst = s;
}

// ---------------------------------------------------------------------------
// Kernel 1: double-buffered async-staged f32 WMMA GEMM fused with per-(row,
// split) top-5 selection. grid = (NTEST/ROWS, NSPLIT); block = 256 = 8 waves.
// Each wave owns a 32x16 column slice of the 32x128 tile (2 accumulators).
// ---------------------------------------------------------------------------
__global__ __launch_bounds__(256) void knn_gemm_topk(const float* __restrict__ xtest,
                                                     const float* __restrict__ xtrain,
                                                     const float* __restrict__ stest,
                                                     const float* __restrict__ strain,
                                                     float* __restrict__ candV,
                                                     int* __restrict__ candI) {
    extern __shared__ float smem[];   // no static LDS -> dynamic region at offset 0

    const int tid  = threadIdx.x;
    const int lane = tid & 31;
    const int wv   = tid >> 5;        // wave id 0..7 -> 16-column slice
    const int lo   = lane & 15;       // row-within-16 / column-within-16
    const int hi   = lane >> 4;       // selects K pair (A/B) or M half (C/D)
    const int rowBase   = blockIdx.x * ROWS;
    const int splitBase = blockIdx.y * SPLIT;

    float* sD   = smem + SD_OFF;
    float* topV = smem + TOPV_OFF;
    int*   topI = (int*)(smem + TOPI_OFF);

    float stestReg = 0.f;
    if (tid < ROWS) {
        stestReg = stest[rowBase + tid];
        #pragma unroll
        for (int j = 0; j < 5; ++j) {
            topV[tid * 5 + j] = NEG_INF;
            topI[tid * 5 + j] = 0x7fffffff;
        }
    }

    // Stage chunk (nt, kc) into pipeline buffer `buf` with async b128 copies.
    auto stage = [&](int buf, int nt, int kc) {
        const unsigned bufB  = (unsigned)(buf * FRAME) * 4u;
        const unsigned kbyte = (unsigned)(kc * KC) * 4u;
        // A tile: 32 rows x 14 float4
        for (int i = tid; i < ROWS * (KC / 4); i += 256) {
            const int r = i / (KC / 4), q = i - r * (KC / 4);
            async_ld128(bufB + (unsigned)(r * LDS_S + q * 4) * 4u, xtest,
                        (unsigned)(rowBase + r) * (DIM * 4u) + kbyte + (unsigned)q * 16u);
        }
        // B tile: 128 rows x 14 float4; tail rows clamped to a valid train row
        // (their distances are never inserted: guarded by nl < SPLIT).
        for (int i = tid; i < COLS * (KC / 4); i += 256) {
            const int r = i / (KC / 4), q = i - r * (KC / 4);
            int nl = nt * COLS + r;
            if (nl >= SPLIT) nl = SPLIT - 1;
            async_ld128(bufB + (unsigned)(SB_OFF + r * LDS_S + q * 4) * 4u, xtrain,
                        (unsigned)(splitBase + nl) * (DIM * 4u) + kbyte + (unsigned)q * 16u);
        }
    };

    // Pipeline prologue: chunk 0 into buffer 0.
    stage(0, 0, 0);
    wait_async0();
    __syncthreads();

    v8f acc0, acc1;
    #pragma unroll
    for (int j = 0; j < 8; ++j) { acc0[j] = 0.f; acc1[j] = 0.f; }

    int nt = 0, kc = 0;
    for (int t = 0; t < NCHUNKS; ++t) {
        const int buf = t & 1;

        // Issue async staging for chunk t+1 into the other buffer.
        if (t + 1 < NCHUNKS) {
            int ntn = nt, kcn = kc + 1;
            if (kcn == NKC) { kcn = 0; ++ntn; }
            stage(buf ^ 1, ntn, kcn);
        }

        // Compute on current buffer: 14 K-steps, 2 WMMAs each.
        // A(16x4 MxK) & B(4x16 KxN) operand layout: lane reads 2 consecutive
        // K values of row (lane&15), K-pair selected by lane>>4.
        {
            const float* A = smem + buf * FRAME;
            const float* B = A + SB_OFF;
            #pragma unroll
            for (int kk = 0; kk < KC / 4; ++kk) {
                const int koff = kk * 4 + 2 * hi;
                const v2f a0 = *(const v2f*)(A + lo * LDS_S + koff);
                const v2f a1 = *(const v2f*)(A + (16 + lo) * LDS_S + koff);
                const v2f bb = *(const v2f*)(B + (wv * 16 + lo) * LDS_S + koff);
                acc0 = __builtin_amdgcn_wmma_f32_16x16x4_f32(
                    false, a0, false, bb, (short)0, acc0, false, false);
                acc1 = __builtin_amdgcn_wmma_f32_16x16x4_f32(
                    false, a1, false, bb, (short)0, acc1, false, false);
            }
        }

        if (kc == NKC - 1) {
            // n-tile finished: dump distances (C/D layout: VGPR j -> M=j+8*hi,
            // N=lane&15), then 32 threads scan/insert while next-chunk async
            // loads are still in flight.
            #pragma unroll
            for (int j = 0; j < 8; ++j) {
                const int c = wv * 16 + lo;
                sD[(j + 8 * hi) * SD_S + c]      = acc0[j];
                sD[(j + 8 * hi + 16) * SD_S + c] = acc1[j];
            }
            __syncthreads();
            if (tid < ROWS) {
                float* tv = &topV[tid * 5];
                int*   ti = &topI[tid * 5];
                for (int c = 0; c < COLS; ++c) {
                    const int nl = nt * COLS + c;
                    if (nl < SPLIT) {
                        const float v = 2.f * sD[tid * SD_S + c] - stestReg
                                        - strain[splitBase + nl] + (float)DIM;
                        insert5(tv, ti, v, splitBase + nl);
                    }
                }
            }
            #pragma unroll
            for (int j = 0; j < 8; ++j) { acc0[j] = 0.f; acc1[j] = 0.f; }
        }

        // Buffer flip: own async loads landed, everyone done with this chunk.
        wait_async0();
        __syncthreads();

        if (++kc == NKC) { kc = 0; ++nt; }
    }

    if (tid < ROWS) {
        const int base = ((rowBase + tid) * NSPLIT + blockIdx.y) * 5;
        #pragma unroll
        for (int j = 0; j < 5; ++j) {
            candV[base + j] = topV[tid * 5 + j];
            candI[base + j] = topI[tid * 5 + j];
        }
    }
}

// ---------------------------------------------------------------------------
// Kernel 2: merge 8 splits x 5 candidates per row -> global top-k -> vote.
// ---------------------------------------------------------------------------
__global__ __launch_bounds__(256) void knn_vote(const int* __restrict__ ytrain,
                                                const int* __restrict__ kptr,
                                                const float* __restrict__ candV,
                                                const int* __restrict__ candI,
                                                float* __restrict__ out) {
    const int b = blockIdx.x * 256 + threadIdx.x;
    if (b >= NTEST) return;

    float tv[5];
    int   ti[5];
    #pragma unroll
    for (int j = 0; j < 5; ++j) { tv[j] = NEG_INF; ti[j] = 0x7fffffff; }

    for (int s = 0; s < NSPLIT; ++s) {
        const int base = (b * NSPLIT + s) * 5;
        #pragma unroll
        for (int j = 0; j < 5; ++j) insert5(tv, ti, candV[base + j], candI[base + j]);
    }

    int kk = *kptr;
    if (kk > 5) kk = 5;
    if (kk < 1) kk = 1;

    int counts[NCLS];
    #pragma unroll
    for (int c = 0; c < NCLS; ++c) counts[c] = 0;
    for (int j = 0; j < kk; ++j) {
        const int idx = ti[j];
        if (idx >= 0 && idx < NTRAIN) {
            const int lab = ytrain[idx];
            if (lab >= 0 && lab < NCLS) counts[lab]++;
        }
    }
    int best = 0, bc = counts[0];
    #pragma unroll
    for (int c = 1; c < NCLS; ++c) {
        if (counts[c] > bc) { bc = counts[c]; best = c; }  // strict > : smallest class wins ties
    }
    out[b] = (float)best;
}

// ---------------------------------------------------------------------------
extern "C" void kernel_launch(void* const* d_in, const int* in_sizes, int n_in,
                              void* d_out, int out_size, void* d_ws, size_t ws_size,
                              hipStream_t stream) {
    const float* xtest  = (const float*)d_in[0];
    const float* xtrain = (const float*)d_in[1];
    const int*   ytrain = (const int*)d_in[2];
    const int*   kptr   = (const int*)d_in[3];
    float*       out    = (float*)d_out;

    // Workspace layout (floats): stest[2048] | strain[50000] | candV[2048*8*5] | candI[...]
    float* ws     = (float*)d_ws;
    float* stest  = ws;
    float* strain = ws + NTEST;
    float* candV  = ws + NTEST + NTRAIN;
    int*   candI  = (int*)(ws + NTEST + NTRAIN + NTEST * NSPLIT * 5);

    // Kernel 0: row sums (52048 rows, 8 waves/block)
    const int rows = NTEST + NTRAIN;
    knn_rowsums<<<(rows + 7) / 8, 256, 0, stream>>>(xtest, xtrain, stest, strain);

    // Kernel 1: async double-buffered WMMA GEMM + per-split top-5
    dim3 grid(NTEST / ROWS, NSPLIT);
    knn_gemm_topk<<<grid, 256, SMEM_FLOATS * sizeof(float), stream>>>(
        xtest, xtrain, stest, strain, candV, candI);

    // Kernel 2: merge + vote
    knn_vote<<<(NTEST + 255) / 256, 256, 0, stream>>>(ytrain, kptr, candV, candI, out);
}